// ScaledDotProductAttention_74663711474182
// MI455X (gfx1250) — compile-verified
//
#include <hip/hip_runtime.h>
#include <stdint.h>

// Head-axis-softmax attention, fully fused per (b, q-tile, k-tile) across all
// 16 heads. B=2, H=16, L=2048, D=64, fp32 in/out. All matrix math on
// v_wmma_f32_16x16x4_f32 (the fp32 matrix path; matches reference precision).

typedef float v2f __attribute__((ext_vector_type(2)));
typedef float v8f __attribute__((ext_vector_type(8)));

constexpr int B_ = 2, H_ = 16, L_ = 2048, D_ = 64;
constexpr int NK = 2;                    // split k-range across workgroups
constexpr int KT_PER = (L_ / 16) / NK;   // 64 k-tiles per chunk
constexpr float SCALE = 0.125f;          // 1/sqrt(64)

// LDS: per head one 16x16 f32 tile, rows padded to 17 dwords (bank-conflict free
// stage-B reads and stage-C A-fragment reads).
constexpr int SSTR = 17;
constexpr int HBLK = 16 * SSTR;

__global__ __launch_bounds__(256) void attn_headsoftmax_fused(
    const float* __restrict__ Q, const float* __restrict__ K,
    const float* __restrict__ V, const uint8_t* __restrict__ M,
    float* __restrict__ part)
{
  __shared__ float sS[H_ * HBLK];   // raw scores, q-major rows
  __shared__ float sP[H_ * HBLK];   // normalized probs, k-major rows

  const int qt     = blockIdx.x;
  const int kchunk = blockIdx.y;
  const int b      = blockIdx.z;
  const int tid    = threadIdx.x;
  const int wv     = tid >> 5;          // wave 0..7 -> owns heads 2wv, 2wv+1
  const int lane   = tid & 31;
  const int l16    = lane & 15;
  const int half   = lane >> 4;

  // Per-head base rows for this lane (K fragment rows use key = l16; V base starts at key 0).
  const float* kbase[2];
  const float* vbase[2];
#pragma unroll
  for (int hh = 0; hh < 2; ++hh) {
    const int h = 2 * wv + hh;
    kbase[hh] = K + (((size_t)(b * H_ + h)) * L_ + l16) * D_;
    vbase[hh] = V + (((size_t)(b * H_ + h)) * L_) * D_;
  }

  // ---- Hoist Q fragments (A-matrix 16x4 f32, 16 substeps over D=64).
  // Lane layout: A[m = lane%16][k = 2*(lane/16) + {0,1}] -> contiguous float2.
  v2f qa[2][16];
#pragma unroll
  for (int hh = 0; hh < 2; ++hh) {
    const int h = 2 * wv + hh;
    const float* qrow = Q + (((size_t)(b * H_ + h)) * L_ + qt * 16 + l16) * D_;
#pragma unroll
    for (int s = 0; s < 16; ++s)
      qa[hh][s] = *reinterpret_cast<const v2f*>(qrow + s * 4 + 2 * half);
  }

  v8f oacc[2][4];
#pragma unroll
  for (int hh = 0; hh < 2; ++hh)
#pragma unroll
    for (int dt = 0; dt < 4; ++dt)
      oacc[hh][dt] = {};

  for (int kti = 0; kti < KT_PER; ++kti) {
    const int kt = kchunk * KT_PER + kti;

    // ---- Stage A: batch-load ALL K fragments for both heads first, then run
    //      the two 16-WMMA chains back-to-back (one load-latency exposure).
    v2f kb[2][16];
#pragma unroll
    for (int hh = 0; hh < 2; ++hh) {
      const float* krow = kbase[hh] + (size_t)kt * 16 * D_;
#pragma unroll
      for (int s = 0; s < 16; ++s)
        kb[hh][s] = *reinterpret_cast<const v2f*>(krow + s * 4 + 2 * half);
    }
    // Prefetch next k-tile's K rows (uniform branch; EXEC stays all-ones).
    if (kti + 1 < KT_PER) {
#pragma unroll
      for (int hh = 0; hh < 2; ++hh) {
        const float* krow_n = kbase[hh] + (size_t)(kt + 1) * 16 * D_;
        __builtin_prefetch(krow_n, 0, 0);
        __builtin_prefetch(krow_n + 32, 0, 0);   // second 128B of the 256B row
      }
    }
#pragma unroll
    for (int hh = 0; hh < 2; ++hh) {
      const int h = 2 * wv + hh;
      v8f acc = {};
#pragma unroll
      for (int s = 0; s < 16; ++s)
        acc = __builtin_amdgcn_wmma_f32_16x16x4_f32(
            false, qa[hh][s], false, kb[hh][s], (short)0, acc, false, false);
#pragma unroll
      for (int r = 0; r < 8; ++r)
        sS[h * HBLK + (r + 8 * half) * SSTR + l16] = acc[r];
    }
    __syncthreads();

    // ---- Stage B: cross-head softmax, one (q,k) element per thread ----
    {
      const int ql = tid >> 4, kl = tid & 15;
      const int qg = qt * 16 + ql, kg = kt * 16 + kl;
      const uint8_t* mp = M + ((size_t)b * H_ * L_ + qg) * L_ + kg;
      float sv[H_];
      float mmax = -3.4e38f;
#pragma unroll
      for (int h = 0; h < H_; ++h) {
        float s = sS[h * HBLK + ql * SSTR + kl] * SCALE;
        uint8_t mk = mp[(size_t)h * L_ * L_];
        s = mk ? -1e9f : s;             // jnp.where(mask, -1e9, score*scale)
        sv[h] = s;
        mmax = fmaxf(mmax, s);
      }
      float sum = 0.f;
#pragma unroll
      for (int h = 0; h < H_; ++h) { sv[h] = __expf(sv[h] - mmax); sum += sv[h]; }
      const float rs = 1.0f / sum;      // all-masked column -> uniform 1/16
#pragma unroll
      for (int h = 0; h < H_; ++h)
        sP[h * HBLK + kl * SSTR + ql] = sv[h] * rs;   // k-major for stage C
    }
    __syncthreads();

    // ---- Stage C: O += P * V; batch A (LDS) + B (V) fragment loads, then
    //      16 back-to-back WMMAs per head.
#pragma unroll
    for (int hh = 0; hh < 2; ++hh) {
      const int h = 2 * wv + hh;
      const float* vrow = vbase[hh] + (size_t)kt * 16 * D_;
      v2f af[4];
      float vx[4][4], vy[4][4];
#pragma unroll
      for (int k4 = 0; k4 < 4; ++k4) {
        const int kk = k4 * 4 + 2 * half;
        af[k4].x = sP[h * HBLK + (kk + 0) * SSTR + l16];
        af[k4].y = sP[h * HBLK + (kk + 1) * SSTR + l16];
#pragma unroll
        for (int dt = 0; dt < 4; ++dt) {
          vx[k4][dt] = vrow[(size_t)(kk + 0) * D_ + dt * 16 + l16];
          vy[k4][dt] = vrow[(size_t)(kk + 1) * D_ + dt * 16 + l16];
        }
      }
      // Prefetch next k-tile's V rows for this head.
      if (kti + 1 < KT_PER) {
        const float* vrow_n = vbase[hh] + ((size_t)(kt + 1) * 16 + l16) * D_;
        __builtin_prefetch(vrow_n, 0, 0);
        __builtin_prefetch(vrow_n + 32, 0, 0);
      }
#pragma unroll
      for (int k4 = 0; k4 < 4; ++k4)
#pragma unroll
        for (int dt = 0; dt < 4; ++dt) {
          v2f bf;
          bf.x = vx[k4][dt];
          bf.y = vy[k4][dt];
          oacc[hh][dt] = __builtin_amdgcn_wmma_f32_16x16x4_f32(
              false, af[k4], false, bf, (short)0, oacc[hh][dt], false, false);
        }
    }
    __syncthreads();
  }

  // ---- Write per-k-chunk partial outputs (coalesced over d) ----
#pragma unroll
  for (int hh = 0; hh < 2; ++hh) {
    const int h = 2 * wv + hh;
#pragma unroll
    for (int dt = 0; dt < 4; ++dt)
#pragma unroll
      for (int r = 0; r < 8; ++r) {
        const int q = qt * 16 + r + 8 * half;
        const int d = dt * 16 + l16;
        part[((((size_t)kchunk * B_ + b) * H_ + h) * L_ + q) * D_ + d] =
            oacc[hh][dt][r];
      }
  }
}

// Deterministic combine of the NK k-chunk partials (no FP atomics).
__global__ __launch_bounds__(256) void reduce_partials(
    const float* __restrict__ part, float* __restrict__ out, int n)
{
  int i = blockIdx.x * 256 + threadIdx.x;
  if (i < n) out[i] = part[i] + part[(size_t)n + i];
}

extern "C" void kernel_launch(void* const* d_in, const int* in_sizes, int n_in,
                              void* d_out, int out_size, void* d_ws, size_t ws_size,
                              hipStream_t stream) {
  const float*   Q = (const float*)d_in[0];
  const float*   K = (const float*)d_in[1];
  const float*   V = (const float*)d_in[2];
  const uint8_t* M = (const uint8_t*)d_in[3];   // jax bool -> 1 byte/elem
  float* out  = (float*)d_out;
  float* part = (float*)d_ws;                   // [NK][B,H,L,D] f32 = 33.6 MB

  dim3 grid(L_ / 16, NK, B_);                   // 512 workgroups x 8 waves
  attn_headsoftmax_fused<<<grid, 256, 0, stream>>>(Q, K, V, M, part);

  const int n = B_ * H_ * L_ * D_;
  reduce_partials<<<(n + 255) / 256, 256, 0, stream>>>(part, out, n);
}